// GraphUNet_51384988729803
// MI455X (gfx1250) — compile-verified
//
#include <hip/hip_runtime.h>
#include <math.h>

#define N_NODES 16384
#define N_EDGES 262144
#define NHR     65536
#define EHR     1048576
#define BATCH   8
#define CP      20            // padded channel stride (max C = 18)
#define KSTEPS  5             // CP / 4 K-steps per WMMA tile

// per-batch workspace stride, in floats
#define PBS (160LL * N_NODES)

// per-batch float offsets inside workspace
#define OFF_FB(i)  ((long long)(i) * N_NODES * CP)                 // 7 feature bufs [N][CP]
#define OFF_M(j)   (7LL*N_NODES*CP + (long long)(j)*N_NODES)       // masks m1..m4
#define OFF_DEG    (7LL*N_NODES*CP + 4LL*N_NODES)
#define OFF_NORM   (7LL*N_NODES*CP + 5LL*N_NODES)
#define OFF_SCORE  (7LL*N_NODES*CP + 6LL*N_NODES)
#define OFF_HR0    (7LL*N_NODES*CP + 7LL*N_NODES)                  // NHR = 4N
#define OFF_DEGHR  (7LL*N_NODES*CP + 11LL*N_NODES)                 // deg, then norm in-place
#define OFF_AGGHR  (7LL*N_NODES*CP + 15LL*N_NODES)
#define OFF_CTRL   (7LL*N_NODES*CP + 19LL*N_NODES)                 // uint ctrl block
#define OFF_SUM    (OFF_CTRL + 16)
// batch-shared normalized pool vectors (after all per-batch regions)
#define OFF_PN0    (8LL*PBS + 0)
#define OFF_PN1    (8LL*PBS + 16)
#define OFF_PN2    (8LL*PBS + 128)
#define OFF_PN3    (8LL*PBS + 336)

typedef float v2f __attribute__((ext_vector_type(2)));
typedef float v8f __attribute__((ext_vector_type(8)));

__device__ __forceinline__ unsigned f2key(float f) {
  unsigned u = __float_as_uint(f);
  return (u & 0x80000000u) ? ~u : (u | 0x80000000u);
}

// ---------------------------------------------------------------- utilities
__global__ void k_fill(float* ws, long long off, int n, float v) {
  float* p = ws + (long long)blockIdx.y * PBS + off;
  int i = blockIdx.x * blockDim.x + threadIdx.x;
  if (i < n) p[i] = v;
}

__global__ void k_copy_x(float* ws, const float* x) {
  int n = blockIdx.x * blockDim.x + threadIdx.x;
  if (n < N_NODES)
    (ws + (long long)blockIdx.y * PBS + OFF_FB(4))[(long long)n * CP] =
        x[(long long)blockIdx.y * N_NODES + n];
}

// ------------------------------------------------------------- graph pieces
__global__ void k_count_deg(float* ws, const int* src, const int* dst, int ne,
                            long long off_deg, long long off_mask) {
  long long b = blockIdx.y;
  const int* s = src + b * (long long)ne;
  const int* d = dst + b * (long long)ne;
  float* deg = ws + b * PBS + off_deg;
  const float* m = (off_mask >= 0) ? (ws + b * PBS + off_mask) : nullptr;
  int e = blockIdx.x * blockDim.x + threadIdx.x;
  if (e < ne) {
    int si = s[e], di = d[e];
    if (!m || (m[si] > 0.5f && m[di] > 0.5f)) atomicAdd(&deg[di], 1.0f);
  }
}

__global__ void k_norm(float* ws, long long off_deg, long long off_norm, int n) {
  long long b = blockIdx.y;
  const float* deg = ws + b * PBS + off_deg;
  float* nrm = ws + b * PBS + off_norm;
  int i = blockIdx.x * blockDim.x + threadIdx.x;
  if (i < n) {
    float dg = deg[i];
    nrm[i] = dg > 0.f ? rsqrtf(fmaxf(dg, 1.f)) : 0.f;
  }
}

__global__ void k_scatter(float* ws, const int* src, const int* dst, int ne,
                          long long off_in1, long long off_in2, long long off_norm,
                          long long off_mask, long long off_agg, int cin, int stride) {
  long long b = blockIdx.y;
  const int* s = src + b * (long long)ne;
  const int* d = dst + b * (long long)ne;
  const float* in1 = ws + b * PBS + off_in1;
  const float* in2 = (off_in2 >= 0) ? ws + b * PBS + off_in2 : nullptr;
  const float* nrm = ws + b * PBS + off_norm;
  const float* m   = (off_mask >= 0) ? ws + b * PBS + off_mask : nullptr;
  float* agg = ws + b * PBS + off_agg;
  int e = blockIdx.x * blockDim.x + threadIdx.x;
  if (e < ne) {
    // prefetch the next tile's source feature row (global_prefetch_b8)
    if (e + (int)blockDim.x < ne) {
      int sn = s[e + blockDim.x];
      __builtin_prefetch(in1 + (long long)sn * stride, 0, 0);
    }
    int si = s[e], di = d[e];
    if (!m || (m[si] > 0.5f && m[di] > 0.5f)) {
      float w = nrm[si];
      if (w != 0.f) {
        const float* r1 = in1 + (long long)si * stride;
        const float* r2 = in2 ? in2 + (long long)si * stride : nullptr;
        float* o = agg + (long long)di * stride;
        for (int c = 0; c < cin; c++) {
          float v = r1[c];
          if (r2) v += r2[c];
          atomicAdd(&o[c], v * w);
        }
      }
    }
  }
}

// out[n][co] = (sum_ci agg[n][ci]*norm[n]*W[ci][co]) + b[co], times optional mask.
// One wave per 16-node tile; V_WMMA_F32_16X16X4_F32, K padded to CP=20.
// Weights are staged zero-padded into LDS so the WMMA loop is branch-free.
__global__ void __launch_bounds__(128)
k_transform(float* ws, long long off_agg, long long off_norm,
            const float* __restrict__ W, const float* __restrict__ bias,
            int cin, int cout, long long off_mask, long long off_out) {
  __shared__ float Wp[CP][32];   // zero-padded weights [K][col]
  __shared__ float Bp[32];       // zero-padded bias

  for (int i = threadIdx.x; i < CP * 32; i += blockDim.x) {
    int k = i >> 5, c = i & 31;
    Wp[k][c] = (k < cin && c < cout) ? W[k * cout + c] : 0.f;
  }
  if (threadIdx.x < 32)
    Bp[threadIdx.x] = ((int)threadIdx.x < cout) ? bias[threadIdx.x] : 0.f;
  __syncthreads();

  long long b = blockIdx.y;
  const float* agg = ws + b * PBS + off_agg;
  const float* nrm = ws + b * PBS + off_norm;
  const float* msk = (off_mask >= 0) ? ws + b * PBS + off_mask : nullptr;
  float* out = ws + b * PBS + off_out;

  int lane = threadIdx.x & 31;
  int wv   = threadIdx.x >> 5;
  int tileBase = (blockIdx.x * 4 + wv) * 16;
  int half = lane >> 4;      // K half: lanes 0-15 -> K{0,1}, 16-31 -> K{2,3}
  int r    = lane & 15;      // A row / B column within tile

  int nodeA = tileBase + r;
  float nr = nrm[nodeA];
  const float* arow = agg + (long long)nodeA * CP;

  v8f acc0 = {0.f, 0.f, 0.f, 0.f, 0.f, 0.f, 0.f, 0.f};
  v8f acc1 = acc0;
#pragma unroll
  for (int s = 0; s < KSTEPS; s++) {
    int k0 = 4 * s + 2 * half;
    v2f a = *(const v2f*)(arow + k0);   // 8B-aligned (even k0, CP even)
    a.x *= nr;
    a.y *= nr;
    v2f b0, b1;
    b0.x = Wp[k0][r];       b0.y = Wp[k0 + 1][r];
    b1.x = Wp[k0][r + 16];  b1.y = Wp[k0 + 1][r + 16];
    acc0 = __builtin_amdgcn_wmma_f32_16x16x4_f32(false, a, false, b0, (short)0, acc0, false, false);
    acc1 = __builtin_amdgcn_wmma_f32_16x16x4_f32(false, a, false, b1, (short)0, acc1, false, false);
  }
#pragma unroll
  for (int t = 0; t < 2; t++) {
    int col = r + 16 * t;
    if (col < cout) {
      v8f acc = t ? acc1 : acc0;
      float bv = Bp[col];
#pragma unroll
      for (int v = 0; v < 8; v++) {
        int node = tileBase + v + 8 * half;   // D layout: vgpr=M row, lane=N col
        float mval = msk ? msk[node] : 1.f;
        out[(long long)node * CP + col] = (acc[v] + bv) * mval;
      }
    }
  }
}

// ----------------------------------------------------------------- pooling
__global__ void k_pnorm(const float* p, int d, float* ws, long long off_pn) {
  __shared__ float red[64];
  int n = d * d;
  float acc = 0.f;
  for (int i = threadIdx.x; i < n; i += blockDim.x) acc += p[i] * p[i];
  red[threadIdx.x] = acc;
  __syncthreads();
  for (int s = 32; s > 0; s >>= 1) {
    if ((int)threadIdx.x < s) red[threadIdx.x] += red[threadIdx.x + s];
    __syncthreads();
  }
  float inv = rsqrtf(red[0]);
  for (int i = threadIdx.x; i < n; i += blockDim.x) (ws + off_pn)[i] = p[i] * inv;
}

__global__ void k_score(float* ws, long long off_x, long long off_pn, int d,
                        long long off_mprev, long long off_score) {
  long long b = blockIdx.y;
  const float* x  = ws + b * PBS + off_x;
  const float* pn = ws + off_pn;                 // batch-shared
  const float* m  = (off_mprev >= 0) ? ws + b * PBS + off_mprev : nullptr;
  float* sc = ws + b * PBS + off_score;
  int n = blockIdx.x * blockDim.x + threadIdx.x;
  if (n < N_NODES) {
    float v;
    if (m && m[n] <= 0.5f) v = -INFINITY;
    else {
      float y0 = 0.f;
      const float* xr = x + (long long)n * CP;
      for (int i = 0; i < d; i++) y0 += xr[i] * pn[i * d + 0];
      v = y0;
    }
    sc[n] = v;
  }
}

// Exact descending top-K threshold via 4-pass byte radix select.
__global__ void k_radix_select(float* ws, long long off_score, int K) {
  long long b = blockIdx.y;
  const float* sc = ws + b * PBS + off_score;
  unsigned* ctrl = (unsigned*)(ws + b * PBS + OFF_CTRL);
  __shared__ unsigned hist[256];
  __shared__ unsigned s_prefix, s_rem;
  if (threadIdx.x == 0) { s_prefix = 0u; s_rem = (unsigned)K; }
  __syncthreads();
  for (int p = 3; p >= 0; --p) {
    for (int i = threadIdx.x; i < 256; i += blockDim.x) hist[i] = 0u;
    __syncthreads();
    unsigned prefix = s_prefix;
    unsigned hmask = (p == 3) ? 0u : (0xFFFFFFFFu << ((p + 1) * 8));
    for (int i = threadIdx.x; i < N_NODES; i += blockDim.x) {
      unsigned k = f2key(sc[i]);
      if ((k & hmask) == (prefix & hmask))
        atomicAdd(&hist[(k >> (p * 8)) & 0xFFu], 1u);
    }
    __syncthreads();
    if (threadIdx.x == 0) {
      unsigned rem = s_rem;
      int bsel = 255;
      while (bsel > 0) {
        unsigned c = hist[bsel];
        if (c >= rem) break;
        rem -= c;
        bsel--;
      }
      s_prefix = prefix | ((unsigned)bsel << (p * 8));
      s_rem = rem;
    }
    __syncthreads();
  }
  if (threadIdx.x == 0) { ctrl[0] = s_prefix; ctrl[1] = s_rem; ctrl[2] = 0u; }
}

__global__ void k_make_mask(float* ws, long long off_score, long long off_mnew) {
  long long b = blockIdx.y;
  const float* sc = ws + b * PBS + off_score;
  unsigned* ctrl = (unsigned*)(ws + b * PBS + OFF_CTRL);
  float* mn = ws + b * PBS + off_mnew;
  int n = blockIdx.x * blockDim.x + threadIdx.x;
  if (n < N_NODES) {
    unsigned key = f2key(sc[n]);
    unsigned T = ctrl[0], rem = ctrl[1];
    float m = 0.f;
    if (key > T) m = 1.f;
    else if (key == T) {
      unsigned pos = atomicAdd(&ctrl[2], 1u);
      if (pos < rem) m = 1.f;
    }
    mn[n] = m;
  }
}

__global__ void k_gate(float* ws, long long off_x, long long off_pn, int d,
                       long long off_mnew, long long off_out) {
  long long b = blockIdx.y;
  const float* x  = ws + b * PBS + off_x;
  const float* pn = ws + off_pn;
  const float* mn = ws + b * PBS + off_mnew;
  float* out = ws + b * PBS + off_out;
  int n = blockIdx.x * blockDim.x + threadIdx.x;
  if (n < N_NODES) {
    const float* xr = x + (long long)n * CP;
    float* orow = out + (long long)n * CP;
    float mv = mn[n];
    for (int c = 0; c < d; c++) {
      float y = 0.f;
      for (int i = 0; i < d; i++) y += xr[i] * pn[i * d + c];
      float sg = 1.f / (1.f + expf(-y));
      orow[c] = xr[c] * sg * mv;
    }
  }
}

// ----------------------------------------------------------------- HR path
__global__ void k_repeat(float* ws) {
  long long b = blockIdx.y;
  const float* u = ws + b * PBS + OFF_FB(5);   // u4 lives in buf5, channel 0
  float* hr = ws + b * PBS + OFF_HR0;
  int i = blockIdx.x * blockDim.x + threadIdx.x;
  if (i < NHR) hr[i] = u[(long long)(i >> 2) * CP];
}

__global__ void k_hr_final(float* ws, const float* W9, const float* b9, float* dout) {
  long long b = blockIdx.y;
  const float* agg = ws + b * PBS + OFF_AGGHR;
  const float* nrm = ws + b * PBS + OFF_DEGHR;   // norm computed in-place
  float* sum = ws + b * PBS + OFF_SUM;
  __shared__ float red[256];
  int i = blockIdx.x * blockDim.x + threadIdx.x;
  float v = 0.f;
  if (i < NHR) {
    v = agg[i] * nrm[i] * W9[0] + b9[0];
    v = v > 0.f ? v : 0.f;
    dout[b * (long long)NHR + i] = v;
  }
  red[threadIdx.x] = v;
  __syncthreads();
  for (int s = 128; s > 0; s >>= 1) {
    if ((int)threadIdx.x < s) red[threadIdx.x] += red[threadIdx.x + s];
    __syncthreads();
  }
  if (threadIdx.x == 0) atomicAdd(sum, red[0]);
}

__global__ void k_mlp(float* ws,
                      const float* W0, const float* W1, const float* W2,
                      const float* W3, const float* W4, const float* W5,
                      const float* B0, const float* B1, const float* B2,
                      const float* B3, const float* B4, const float* B5,
                      float* dout) {
  long long b = blockIdx.y;
  __shared__ float h[2][64];
  if (threadIdx.x == 0) h[0][0] = ws[b * PBS + OFF_SUM] / (float)NHR;
  __syncthreads();
  const float* Ws[6] = {W0, W1, W2, W3, W4, W5};
  const float* Bs[6] = {B0, B1, B2, B3, B4, B5};
  const int din[6] = {1, 20, 40, 30, 10, 5};
  const int dop[6] = {20, 40, 30, 10, 5, 3};
  int cur = 0;
  for (int l = 0; l < 6; l++) {
    int nxt = cur ^ 1;
    if ((int)threadIdx.x < dop[l]) {
      float a = Bs[l][threadIdx.x];
      for (int i = 0; i < din[l]; i++) a += h[cur][i] * Ws[l][i * dop[l] + threadIdx.x];
      if (l < 5) { a = (a >= 0.f) ? a : -0.8f * a; a = tanhf(a); }
      h[nxt][threadIdx.x] = a;
    }
    __syncthreads();
    cur = nxt;
  }
  if (threadIdx.x < 3)
    dout[(long long)BATCH * NHR + b * 3 + threadIdx.x] = h[cur][threadIdx.x];
}

// ------------------------------------------------------------------ driver
extern "C" void kernel_launch(void* const* d_in, const int* in_sizes, int n_in,
                              void* d_out, int out_size, void* d_ws, size_t ws_size,
                              hipStream_t stream) {
  const float* x      = (const float*)d_in[0];
  const int*   src    = (const int*)d_in[1];
  const int*   dst    = (const int*)d_in[2];
  const int*   src_hr = (const int*)d_in[3];
  const int*   dst_hr = (const int*)d_in[4];
  const float *convW[10], *convB[10], *poolP[4], *projW[6], *projB[6];
  for (int i = 0; i < 10; i++) { convW[i] = (const float*)d_in[5 + i];  convB[i] = (const float*)d_in[15 + i]; }
  for (int i = 0; i < 4;  i++) poolP[i] = (const float*)d_in[25 + i];
  for (int i = 0; i < 6;  i++) { projW[i] = (const float*)d_in[29 + i]; projB[i] = (const float*)d_in[35 + i]; }
  float* out = (float*)d_out;
  float* ws  = (float*)d_ws;

  dim3 blk(256);
  dim3 grdN((N_NODES + 255) / 256, BATCH);
  dim3 grdE((N_EDGES + 255) / 256, BATCH);
  dim3 grdHR((NHR + 255) / 256, BATCH);
  dim3 grdEHR((EHR + 255) / 256, BATCH);
  dim3 grdT(N_NODES / 64, BATCH), blkT(128);

  // batch-shared normalized pool projection vectors
  const long long offPN[4] = {OFF_PN0, OFF_PN1, OFF_PN2, OFF_PN3};
  const int pd[4] = {1, 10, 14, 18};
  for (int i = 0; i < 4; i++)
    k_pnorm<<<dim3(1, 1), 64, 0, stream>>>(poolP[i], pd[i], ws, offPN[i]);

  k_copy_x<<<grdN, blk, 0, stream>>>(ws, x);

  auto gconv = [&](long long in1, long long in2, long long emask,
                   int cin, int cout, int wi, long long omask, long long outb) {
    k_fill<<<dim3((N_NODES + 255) / 256, BATCH), blk, 0, stream>>>(ws, OFF_DEG, N_NODES, 0.f);
    k_fill<<<dim3((N_NODES * CP + 255) / 256, BATCH), blk, 0, stream>>>(ws, OFF_FB(6), N_NODES * CP, 0.f);
    k_count_deg<<<grdE, blk, 0, stream>>>(ws, src, dst, N_EDGES, OFF_DEG, emask);
    k_norm<<<grdN, blk, 0, stream>>>(ws, OFF_DEG, OFF_NORM, N_NODES);
    k_scatter<<<grdE, blk, 0, stream>>>(ws, src, dst, N_EDGES, in1, in2, OFF_NORM, emask, OFF_FB(6), cin, CP);
    k_transform<<<grdT, blkT, 0, stream>>>(ws, OFF_FB(6), OFF_NORM, convW[wi], convB[wi], cin, cout, omask, outb);
  };
  auto pool = [&](long long xin, int d, long long pnOff, long long mprev, int K, long long mnew) {
    k_score<<<grdN, blk, 0, stream>>>(ws, xin, pnOff, d, mprev, OFF_SCORE);
    k_radix_select<<<dim3(1, BATCH), 256, 0, stream>>>(ws, OFF_SCORE, K);
    k_make_mask<<<grdN, blk, 0, stream>>>(ws, OFF_SCORE, mnew);
    k_gate<<<grdN, blk, 0, stream>>>(ws, xin, pnOff, d, mnew, OFF_FB(4));
  };

  const int K1 = 12288, K2 = 9216, K3 = 6912, K4 = 5184;

  // down path
  gconv(OFF_FB(4), -1, -1, 1, 1, 0, -1, OFF_FB(0));                   // h0
  pool(OFF_FB(0), 1, OFF_PN0, -1, K1, OFF_M(0));                      // x1, m1
  gconv(OFF_FB(4), -1, OFF_M(0), 1, 10, 1, OFF_M(0), OFF_FB(1));      // d1
  pool(OFF_FB(1), 10, OFF_PN1, OFF_M(0), K2, OFF_M(1));               // x2, m2
  gconv(OFF_FB(4), -1, OFF_M(1), 10, 14, 2, OFF_M(1), OFF_FB(2));     // d2
  pool(OFF_FB(2), 14, OFF_PN2, OFF_M(1), K3, OFF_M(2));               // x3, m3
  gconv(OFF_FB(4), -1, OFF_M(2), 14, 18, 3, OFF_M(2), OFF_FB(3));     // d3
  pool(OFF_FB(3), 18, OFF_PN3, OFF_M(2), K4, OFF_M(3));               // x4, m4
  gconv(OFF_FB(4), -1, OFF_M(3), 18, 18, 4, OFF_M(3), OFF_FB(5));     // bn -> buf5
  // up path with skips
  gconv(OFF_FB(5), OFF_FB(3), OFF_M(2), 18, 14, 5, OFF_M(2), OFF_FB(4)); // u1
  gconv(OFF_FB(4), OFF_FB(2), OFF_M(1), 14, 10, 6, OFF_M(1), OFF_FB(5)); // u2
  gconv(OFF_FB(5), OFF_FB(1), OFF_M(0), 10, 1, 7, OFF_M(0), OFF_FB(4)); // u3
  gconv(OFF_FB(4), OFF_FB(0), -1, 1, 1, 8, -1, OFF_FB(5));              // u4 -> buf5

  // HR broadcast + conv + mean
  k_repeat<<<grdHR, blk, 0, stream>>>(ws);
  k_fill<<<dim3((NHR + 255) / 256, BATCH), blk, 0, stream>>>(ws, OFF_DEGHR, NHR, 0.f);
  k_fill<<<dim3((NHR + 255) / 256, BATCH), blk, 0, stream>>>(ws, OFF_AGGHR, NHR, 0.f);
  k_count_deg<<<grdEHR, blk, 0, stream>>>(ws, src_hr, dst_hr, EHR, OFF_DEGHR, -1);
  k_norm<<<grdHR, blk, 0, stream>>>(ws, OFF_DEGHR, OFF_DEGHR, NHR);
  k_scatter<<<grdEHR, blk, 0, stream>>>(ws, src_hr, dst_hr, EHR, OFF_HR0, -1, OFF_DEGHR, -1, OFF_AGGHR, 1, 1);
  k_fill<<<dim3(1, BATCH), blk, 0, stream>>>(ws, OFF_SUM, 1, 0.f);
  k_hr_final<<<grdHR, 256, 0, stream>>>(ws, convW[9], convB[9], out);

  // projection MLP
  k_mlp<<<dim3(1, BATCH), 64, 0, stream>>>(ws,
      projW[0], projW[1], projW[2], projW[3], projW[4], projW[5],
      projB[0], projB[1], projB[2], projB[3], projB[4], projB[5], out);
}